// GNN_SAGE_37821482008647
// MI455X (gfx1250) — compile-verified
//
#include <hip/hip_runtime.h>
#include <hip/hip_bf16.h>

typedef __attribute__((ext_vector_type(2))) float v2f;
typedef __attribute__((ext_vector_type(8))) float v8f;

#define DD       128
#define NNODES   50000
#define NEDGES   600000
#define NLAYERS  3
#define NGRAPHS  512
#define MTILES   (NNODES / 16)      // 3125, exact
#define WPB      8                  // waves per block (256 threads)
#define LSTRIDE  33                 // padded LDS row stride (K-chunk = 32)

// ---------------------------------------------------------------- utilities
__global__ void zero_kernel(float* __restrict__ p, int n) {
    int i = blockIdx.x * blockDim.x + threadIdx.x;
    if (i < n) p[i] = 0.0f;
}

__global__ void deg_kernel(const int* __restrict__ dst, float* __restrict__ deg) {
    int e = blockIdx.x * blockDim.x + threadIdx.x;
    if (e < NEDGES) atomicAdd(&deg[dst[e]], 1.0f);
}

// ------------------------------------------------------- edge gather/scatter
// one thread per (edge, 4-feature chunk): 32 threads cover one 128-wide row
__global__ __launch_bounds__(256)
void scatter_kernel(const float* __restrict__ x, const int* __restrict__ src,
                    const int* __restrict__ dst, float* __restrict__ agg) {
    long long idx = (long long)blockIdx.x * blockDim.x + threadIdx.x;
    int e = (int)(idx >> 5);
    int q = (int)(idx & 31);
    if (e >= NEDGES) return;
    int s = src[e], t = dst[e];
    const float4 v = *(const float4*)(x + (size_t)s * DD + q * 4);
    float* ap = agg + (size_t)t * DD + q * 4;
    atomicAdd(ap + 0, v.x);
    atomicAdd(ap + 1, v.y);
    atomicAdd(ap + 2, v.z);
    atomicAdd(ap + 3, v.w);
}

// ------------------------------------------------------------- SAGE layer GEMM
// xout = relu( (agg/deg) @ Wl^T + bl + xin @ Wr^T )
// Wave owns one 16-row tile x all 128 cols (8 v8f accumulators).
// V_WMMA_F32_16X16X4_F32; weights staged through LDS in 128x32 K-slices.
// A operands prefetched per K-chunk; B operands batched per K-step so the
// 16 WMMAs per step issue back-to-back instead of one-per-ds_load.
__global__ __launch_bounds__(256)
void sage_gemm_kernel(const float* __restrict__ xin, const float* __restrict__ agg,
                      const float* __restrict__ deg,
                      const float* __restrict__ Wl, const float* __restrict__ bl,
                      const float* __restrict__ Wr, float* __restrict__ xout) {
    __shared__ float sWl[DD * LSTRIDE];
    __shared__ float sWr[DD * LSTRIDE];

    const int tid  = threadIdx.x;
    const int wave = tid >> 5;
    const int lane = tid & 31;
    const int nl   = lane & 15;   // lane within half: M for A, N for B/C
    const int hi   = lane >> 4;   // which K/M half this lane holds

    const int tile    = blockIdx.x * WPB + wave;
    const bool active = (tile < MTILES);
    const int m0   = tile * 16;
    const int mrow = active ? (m0 + nl) : 0;

    const float invd = active ? (1.0f / fmaxf(deg[mrow], 1.0f)) : 0.0f;

    v8f acc[8];
#pragma unroll
    for (int nt = 0; nt < 8; ++nt) acc[nt] = (v8f){0,0,0,0,0,0,0,0};

    for (int kk0 = 0; kk0 < DD; kk0 += 32) {
        __syncthreads();
        // cooperative load of a 128x32 K-slice of each weight matrix (float4)
        for (int t = tid; t < DD * 8; t += 256) {
            int n  = t >> 3;
            int k4 = (t & 7) * 4;
            const float4 wl4 = *(const float4*)(Wl + (size_t)n * DD + kk0 + k4);
            const float4 wr4 = *(const float4*)(Wr + (size_t)n * DD + kk0 + k4);
            float* pl = sWl + n * LSTRIDE + k4;
            float* pr = sWr + n * LSTRIDE + k4;
            pl[0] = wl4.x; pl[1] = wl4.y; pl[2] = wl4.z; pl[3] = wl4.w;
            pr[0] = wr4.x; pr[1] = wr4.y; pr[2] = wr4.z; pr[3] = wr4.w;
        }
        __syncthreads();

        if (active) {
            // ---- prefetch all A operands for this K-chunk (16 x b64 loads)
            v2f ah[8], ax[8];
            const float* hrow = agg + (size_t)mrow * DD + kk0 + 2 * hi;
            const float* xrow = xin + (size_t)mrow * DD + kk0 + 2 * hi;
#pragma unroll
            for (int s = 0; s < 8; ++s) {
                ah[s].x = hrow[4 * s];     ah[s].y = hrow[4 * s + 1];
                ax[s].x = xrow[4 * s];     ax[s].y = xrow[4 * s + 1];
            }
#pragma unroll
            for (int s = 0; s < 8; ++s) { ah[s].x *= invd; ah[s].y *= invd; }

#pragma unroll
            for (int s = 0; s < 8; ++s) {
                const int klh = 4 * s + 2 * hi;   // this lane's K pair in slice
                // ---- batch-load all 16 B operands into distinct registers
                v2f bwl[8], bwr[8];
#pragma unroll
                for (int nt = 0; nt < 8; ++nt) {
                    const int ncol = nt * 16 + nl;
                    bwl[nt].x = sWl[ncol * LSTRIDE + klh];
                    bwl[nt].y = sWl[ncol * LSTRIDE + klh + 1];
                    bwr[nt].x = sWr[ncol * LSTRIDE + klh];
                    bwr[nt].y = sWr[ncol * LSTRIDE + klh + 1];
                }
                // ---- 8 independent accumulate chains, Wl pass then Wr pass
#pragma unroll
                for (int nt = 0; nt < 8; ++nt)
                    acc[nt] = __builtin_amdgcn_wmma_f32_16x16x4_f32(
                        false, ah[s], false, bwl[nt], (short)0, acc[nt], false, false);
#pragma unroll
                for (int nt = 0; nt < 8; ++nt)
                    acc[nt] = __builtin_amdgcn_wmma_f32_16x16x4_f32(
                        false, ax[s], false, bwr[nt], (short)0, acc[nt], false, false);
            }
        }
    }

    if (active) {
#pragma unroll
        for (int nt = 0; nt < 8; ++nt) {
            const int col  = nt * 16 + nl;
            const float bias = bl[col];
#pragma unroll
            for (int r = 0; r < 8; ++r) {
                // C/D layout: VGPR r holds M = r + 8*hi, N = nl
                float v = acc[nt][r] + bias;
                v = fmaxf(v, 0.0f);
                xout[(size_t)(m0 + r + 8 * hi) * DD + col] = v;
            }
        }
    }
}

// ---------------------------------------------------------------- pooling
__global__ __launch_bounds__(256)
void pool_kernel(const float* __restrict__ x, const int* __restrict__ batch,
                 float* __restrict__ pool) {
    long long idx = (long long)blockIdx.x * blockDim.x + threadIdx.x;
    int i = (int)(idx >> 5);
    int q = (int)(idx & 31);
    if (i >= NNODES) return;
    int b = batch[i];
    const float4 v = *(const float4*)(x + (size_t)i * DD + q * 4);
    float* pp = pool + (size_t)b * DD + q * 4;
    atomicAdd(pp + 0, v.x);
    atomicAdd(pp + 1, v.y);
    atomicAdd(pp + 2, v.z);
    atomicAdd(pp + 3, v.w);
}

__global__ void cnt_kernel(const int* __restrict__ batch, float* __restrict__ cnt) {
    int i = blockIdx.x * blockDim.x + threadIdx.x;
    if (i < NNODES) atomicAdd(&cnt[batch[i]], 1.0f);
}

// one wave (32 lanes) per graph: dot(pool[g]/cnt, lin_w) + lin_b
__global__ __launch_bounds__(32)
void out_kernel(const float* __restrict__ pool, const float* __restrict__ cnt,
                const float* __restrict__ lin_w, const float* __restrict__ lin_b,
                float* __restrict__ out) {
    int g = blockIdx.x;
    int lane = threadIdx.x;
    const float* pg = pool + (size_t)g * DD;
    float s = 0.0f;
#pragma unroll
    for (int j = 0; j < 4; ++j) {
        int d = lane * 4 + j;
        s += pg[d] * lin_w[d];
    }
#pragma unroll
    for (int off = 16; off > 0; off >>= 1)
        s += __shfl_xor(s, off, 32);
    if (lane == 0)
        out[g] = s / fmaxf(cnt[g], 1.0f) + lin_b[0];
}

// ---------------------------------------------------------------- launcher
extern "C" void kernel_launch(void* const* d_in, const int* in_sizes, int n_in,
                              void* d_out, int out_size, void* d_ws, size_t ws_size,
                              hipStream_t stream) {
    const float* x     = (const float*)d_in[0];
    const int*   edge  = (const int*)d_in[1];   // [2][E]
    const int*   batch = (const int*)d_in[2];
    const float* Wl    = (const float*)d_in[3]; // [L][D][D]
    const float* bl    = (const float*)d_in[4]; // [L][D]
    const float* Wr    = (const float*)d_in[5]; // [L][D][D]
    const float* lin_w = (const float*)d_in[6]; // [D]
    const float* lin_b = (const float*)d_in[7]; // [1]
    float* out = (float*)d_out;

    const int* src = edge;
    const int* dst = edge + NEDGES;

    // workspace layout (floats)
    float* ws   = (float*)d_ws;
    float* deg  = ws;                               // N
    float* agg  = deg + NNODES;                     // N*D
    float* xa   = agg + (size_t)NNODES * DD;        // N*D
    float* xb   = xa  + (size_t)NNODES * DD;        // N*D
    float* pool = xb  + (size_t)NNODES * DD;        // G*D
    float* cnt  = pool + (size_t)NGRAPHS * DD;      // G

    const int TB = 256;
    const int ND = NNODES * DD;

    // degrees
    zero_kernel<<<(NNODES + TB - 1) / TB, TB, 0, stream>>>(deg, NNODES);
    deg_kernel<<<(NEDGES + TB - 1) / TB, TB, 0, stream>>>(dst, deg);

    const long long ethreads = (long long)NEDGES * 32;
    const int eblocks = (int)((ethreads + TB - 1) / TB);
    const int gemm_blocks = (MTILES + WPB - 1) / WPB;

    const float* xin = x;
    float* xout = xa;
    for (int l = 0; l < NLAYERS; ++l) {
        zero_kernel<<<(ND + TB - 1) / TB, TB, 0, stream>>>(agg, ND);
        scatter_kernel<<<eblocks, TB, 0, stream>>>(xin, src, dst, agg);
        sage_gemm_kernel<<<gemm_blocks, TB, 0, stream>>>(
            xin, agg, deg,
            Wl + (size_t)l * DD * DD, bl + (size_t)l * DD,
            Wr + (size_t)l * DD * DD, xout);
        xin  = xout;
        xout = (xout == xa) ? xb : xa;
    }
    const float* xfin = xin;  // output of layer 2 (== xa)

    // pooling + head
    zero_kernel<<<(NGRAPHS * DD + TB - 1) / TB, TB, 0, stream>>>(pool, NGRAPHS * DD);
    zero_kernel<<<(NGRAPHS + TB - 1) / TB, TB, 0, stream>>>(cnt, NGRAPHS);
    const long long pthreads = (long long)NNODES * 32;
    pool_kernel<<<(int)((pthreads + TB - 1) / TB), TB, 0, stream>>>(xfin, batch, pool);
    cnt_kernel<<<(NNODES + TB - 1) / TB, TB, 0, stream>>>(batch, cnt);
    out_kernel<<<NGRAPHS, 32, 0, stream>>>(pool, cnt, lin_w, lin_b, out);
}